// LinearGaussianTree_21036749816040
// MI455X (gfx1250) — compile-verified
//
#include <hip/hip_runtime.h>
#include <stdint.h>

// ---------------------------------------------------------------------------
// LinearGaussianTree on MI455X (gfx1250).
// Memory-bound streaming op (~536 MB traffic, ~0.1 GFLOP) -> no WMMA applies.
// CDNA5 path used: global_load_async_to_lds_b32 + s_wait_asynccnt to stage
// each chain's noise row into LDS while the dependent tree walk proceeds.
// ---------------------------------------------------------------------------

#define BLOCK   256
#define DEPTHL  12
#define NODES   4095          // 2^12 - 1
#define REST    4094
#define CHAINS  8192          // M_BATCH(128) * NODE_SIZE(64)
#define SCALEF  0.1f

// ---- s_wait_asynccnt: builtin if the toolchain has it, else inline asm ----
#if defined(__has_builtin) && __has_builtin(__builtin_amdgcn_s_wait_asynccnt)
#define WAIT_ASYNC(n) __builtin_amdgcn_s_wait_asynccnt(n)
#else
#define WAIT_ASYNC(n) asm volatile("s_wait_asynccnt %0" ::"n"(n))
#endif

// ---- async global -> LDS, 4 bytes per lane (VDST = per-lane LDS byte addr) --
__device__ __forceinline__ void async_ld_b32(unsigned lds_byte_addr, const float* g)
{
    asm volatile("global_load_async_to_lds_b32 %0, %1, off"
                 :
                 : "v"(lds_byte_addr), "v"((unsigned long long)(uintptr_t)g)
                 : "memory");
}

// One tree level. N = 2^D nodes. Output layout offset: 4096 - 2^(D+1)
// (levels concatenated in reverse), parents at 4096 - 2^D, root at 4094.
// Inputs (weights, noise) use forward offset 2^D - 2.
template <int D>
__device__ __forceinline__ void do_level(const float* __restrict__ weights,
                                         float* s_samp, const float* s_noise,
                                         float* __restrict__ samp_out,
                                         float* __restrict__ ms_out,
                                         float* __restrict__ sc_out,
                                         int tid)
{
    constexpr int N      = 1 << D;
    constexpr int ROUNDS = (2 * N + 253) >> 8;   // async rounds needed so far
    constexpr int OIN    = N - 2;                // weights / noise offset
    constexpr int OOUT   = 4096 - 2 * N;         // this level in output layout
    constexpr int OPAR   = 4096 - N;             // parent level in output layout

    WAIT_ASYNC(16 - ROUNDS);   // first ROUNDS noise rounds have landed in LDS
    __syncthreads();           // publish parent samples + everyone's waits

    for (int j = tid; j < N; j += BLOCK) {
        float wr = weights[OIN + j];
        float e  = __expf(-wr);
        float w  = __builtin_amdgcn_rcpf(1.0f + e);     // sigmoid
        float sc = (1.0f - w) * SCALEF;
        float p  = s_samp[OPAR + (j >> 1)];
        float m  = w * p;
        float s  = fmaf(sc, s_noise[OIN + j], m);
        s_samp[OOUT + j]   = s;
        samp_out[OOUT + j] = s;
        ms_out[OOUT + j]   = m;
        sc_out[OOUT + j]   = sc;
    }
}

__global__ __launch_bounds__(BLOCK) void
lgt_kernel(const float* __restrict__ weights,
           const float* __restrict__ noise_root,
           const float* __restrict__ noise_rest,
           float* __restrict__ out)
{
    __shared__ float s_samp[4096];    // samples, output layout (root at 4094)
    __shared__ float s_noise[4096];   // staged noise row

    const int tid   = threadIdx.x;
    const int chain = blockIdx.x;     // m*64 + k

    const float* nrow = noise_rest + (size_t)chain * REST;
    float* samp_out = out + (size_t)chain * NODES;
    float* ms_out   = out + (size_t)CHAINS * NODES + (size_t)chain * REST;
    float* sc_out   = out + (size_t)CHAINS * (NODES + REST) + (size_t)chain * REST;

    // Kick off async staging of the whole noise row (16 rounds x 256 lanes,
    // 4B each). Every wave issues exactly 16 async ops (last round partially
    // EXEC-masked but still issued), so per-wave ASYNCcnt thresholds are exact.
    {
        const unsigned lds0 = (unsigned)(size_t)(&s_noise[0]);
#pragma unroll
        for (int r = 0; r < 16; ++r) {
            int i = r * BLOCK + tid;
            if (i < REST) async_ld_b32(lds0 + 4u * (unsigned)i, nrow + i);
        }
    }

    // Root: samples index 4094 (last), value = noise_root (IN_SCALE == 1).
    if (tid == 0) {
        float r = noise_root[chain];
        s_samp[NODES - 1]   = r;
        samp_out[NODES - 1] = r;
    }

    do_level<1>(weights, s_samp, s_noise, samp_out, ms_out, sc_out, tid);
    do_level<2>(weights, s_samp, s_noise, samp_out, ms_out, sc_out, tid);
    do_level<3>(weights, s_samp, s_noise, samp_out, ms_out, sc_out, tid);
    do_level<4>(weights, s_samp, s_noise, samp_out, ms_out, sc_out, tid);
    do_level<5>(weights, s_samp, s_noise, samp_out, ms_out, sc_out, tid);
    do_level<6>(weights, s_samp, s_noise, samp_out, ms_out, sc_out, tid);
    do_level<7>(weights, s_samp, s_noise, samp_out, ms_out, sc_out, tid);
    do_level<8>(weights, s_samp, s_noise, samp_out, ms_out, sc_out, tid);
    do_level<9>(weights, s_samp, s_noise, samp_out, ms_out, sc_out, tid);
    do_level<10>(weights, s_samp, s_noise, samp_out, ms_out, sc_out, tid);
    do_level<11>(weights, s_samp, s_noise, samp_out, ms_out, sc_out, tid);
}

extern "C" void kernel_launch(void* const* d_in, const int* in_sizes, int n_in,
                              void* d_out, int out_size, void* d_ws, size_t ws_size,
                              hipStream_t stream)
{
    (void)in_sizes; (void)n_in; (void)d_ws; (void)ws_size; (void)out_size;
    const float* weights    = (const float*)d_in[0];  // (4095,)
    const float* noise_root = (const float*)d_in[1];  // (128,64,1)
    const float* noise_rest = (const float*)d_in[2];  // (128,64,4094)
    float* out = (float*)d_out;                       // samples | ms | scales

    lgt_kernel<<<CHAINS, BLOCK, 0, stream>>>(weights, noise_root, noise_rest, out);
}